// DefConv_386547056931
// MI455X (gfx1250) — compile-verified
//
#include <hip/hip_runtime.h>
#include <math.h>

// ---------------------------------------------------------------------------
// Deformable Conv v2 forward for MI455X (gfx1250), fp32 WMMA path.
//   x:(8,128,80,80) f32, w_off:(27,128,3,3), w_dcn:(128,128,3,3)
//   GEMM formulation: W[M x 1152] @ col[1152 x 51200]
//   Matrix op: v_wmma_f32_16x16x4_f32 (full fp32 precision, wave32)
// Round-2 changes: zero-padded offset weights (no exec-mask dance in the
// inner loop) + 1-deep software pipelining of A/B fragment loads.
// ---------------------------------------------------------------------------

typedef __attribute__((ext_vector_type(2))) float v2f;
typedef __attribute__((ext_vector_type(8))) float v8f;

#define B_     8
#define C_     128
#define H_     80
#define W_     80
#define HW     (H_ * W_)              // 6400
#define N_TOT  (B_ * HW)              // 51200
#define CK     (C_ * 9)               // 1152 (K dim of GEMM, index ci*9+k)
#define TILE_N 64
#define LDS_STRIDE (CK + 2)           // 1154 dwords: pad -> conflict-free b64 reads
#define LDS_BYTES  (TILE_N * LDS_STRIDE * 4)   // 295,424 B (< 320 KB WGP LDS)
#define EPS_   1e-5f

__device__ __forceinline__ float reluf(float v) { return v > 0.f ? v : 0.f; }

// ---------------------------------------------------------------------------
// Kernel 0: zero-pad offset weights 27x1152 -> 32x1152 so the GEMM A-loads
// need no per-lane validity guard (keeps EXEC untouched in the hot loop).
// ---------------------------------------------------------------------------
extern "C" __global__ void __launch_bounds__(256)
prep_weights_kernel(const float* __restrict__ w_off, float* __restrict__ wpad)
{
    const int idx = blockIdx.x * 256 + threadIdx.x;   // < 32*1152 exactly
    const int m = idx / CK;
    wpad[idx] = (m < 27) ? w_off[idx] : 0.f;
}

// ---------------------------------------------------------------------------
// Kernel 1: offset conv.  col = im2col(relu(x)) staged in LDS, then
// 32(M, 27 valid) x 64(N) WMMA GEMM over K=1152.  Epilogue writes channel
// planes om[ch][n]: ch<18 raw (dy/dx interleaved), ch>=18 sigmoid (mask).
// ---------------------------------------------------------------------------
extern "C" __global__ void __launch_bounds__(256)
offset_conv_kernel(const float* __restrict__ x, const float* __restrict__ wpad,
                   const float* __restrict__ b_off, float* __restrict__ om)
{
    extern __shared__ float col[];   // [TILE_N][LDS_STRIDE]
    const int t   = threadIdx.x;
    const int pix = t & 63;          // lanes within a wave have consecutive pix
    const int cig = t >> 6;          // channel group 0..3 (32 channels each)
    const int nb  = blockIdx.x * TILE_N;
    const int n   = nb + pix;
    const int b   = n / HW, hw = n % HW;
    const int h   = hw / W_, w = hw % W_;

    // ---- build col tile: plain shifted relu(x), zero-padded borders ----
    const float* xb = x + (long)b * C_ * HW;
    #pragma unroll
    for (int k = 0; k < 9; ++k) {
        const int iy = h - 1 + k / 3;
        const int ix = w - 1 + k % 3;
        const bool ok = (iy >= 0) & (iy < H_) & (ix >= 0) & (ix < W_);
        const int sp = (ok ? iy : 0) * W_ + (ok ? ix : 0);
        for (int ci = cig * 32; ci < cig * 32 + 32; ++ci) {
            float v = ok ? reluf(xb[ci * HW + sp]) : 0.f;
            col[pix * LDS_STRIDE + ci * 9 + k] = v;
        }
    }
    __syncthreads();

    // ---- WMMA GEMM: wave -> (Mtile = wave&1, Nsub = wave>>1) ----
    const int wave = t >> 5, lane = t & 31;
    const int Mb   = (wave & 1) * 16;
    const int pixb = (wave >> 1) * 16;
    const int lm   = lane & 15;            // row/col within half
    const int lk   = (lane >> 4) * 2;      // lanes 16..31 carry K+2..K+3
    const float* arow = wpad + (long)(Mb + lm) * CK + lk;  // padded: no guard
    const float* brow = col + (pixb + lm) * LDS_STRIDE + lk;

    v8f c = {};
    v2f a, bf;
    a.x  = arow[0]; a.y  = arow[1];
    bf.x = brow[0]; bf.y = brow[1];
    for (int kk = 0; kk < CK - 4; kk += 4) {
        v2f an, bn;                               // prefetch next K-step
        an.x = arow[kk + 4]; an.y = arow[kk + 5];
        bn.x = brow[kk + 4]; bn.y = brow[kk + 5];
        c = __builtin_amdgcn_wmma_f32_16x16x4_f32(
                false, a, false, bf, (short)0, c, false, false);
        a = an; bf = bn;
    }
    c = __builtin_amdgcn_wmma_f32_16x16x4_f32(
            false, a, false, bf, (short)0, c, false, false);

    // ---- epilogue: C/D layout VGPR r -> M = Mb + r + 8*(lane>=16), N = lm ----
    const int hi = lane >> 4;
    const int nn = nb + pixb + lm;
    #pragma unroll
    for (int r = 0; r < 8; ++r) {
        const int ch = Mb + r + 8 * hi;
        if (ch < 27) {
            float v = c[r] + b_off[ch];
            if (ch >= 18) v = 1.f / (1.f + __expf(-v));   // mask = sigmoid
            om[(long)ch * N_TOT + nn] = v;
        }
    }
}

// ---------------------------------------------------------------------------
// Kernel 2: deformable col build (bilinear, per-tap validity, x mask) + main
// 128 x 64 WMMA GEMM (8 waves x one 16-row M tile, 4 N-subtiles each).
// Writes pre-BN result as channel planes out_pre[co][n].
// ---------------------------------------------------------------------------
extern "C" __global__ void __launch_bounds__(256)
dcn_conv_kernel(const float* __restrict__ x, const float* __restrict__ w_dcn,
                const float* __restrict__ b_dcn, const float* __restrict__ om,
                float* __restrict__ out_pre)
{
    extern __shared__ float col[];
    const int t   = threadIdx.x;
    const int pix = t & 63;
    const int cig = t >> 6;
    const int nb  = blockIdx.x * TILE_N;
    const int n   = nb + pix;
    const int b   = n / HW, hw = n % HW;
    const int h   = hw / W_, w = hw % W_;
    const float* xb = x + (long)b * C_ * HW;

    // ---- build deformable col tile ----
    #pragma unroll
    for (int k = 0; k < 9; ++k) {
        const float dy = om[(long)(2 * k)     * N_TOT + n];
        const float dx = om[(long)(2 * k + 1) * N_TOT + n];
        const float mk = om[(long)(18 + k)    * N_TOT + n];
        const float py = (float)(h - 1 + k / 3) + dy;
        const float px = (float)(w - 1 + k % 3) + dx;
        const float fy = floorf(py), fx = floorf(px);
        const int   y0 = (int)fy,    x0 = (int)fx;
        const float ly = py - fy,    lx = px - fx;
        const bool vy0 = (y0 >= 0) & (y0 < H_);
        const bool vy1 = (y0 + 1 >= 0) & (y0 + 1 < H_);
        const bool vx0 = (x0 >= 0) & (x0 < W_);
        const bool vx1 = (x0 + 1 >= 0) & (x0 + 1 < W_);
        const int cy0 = min(max(y0, 0), H_ - 1), cy1 = min(max(y0 + 1, 0), H_ - 1);
        const int cx0 = min(max(x0, 0), W_ - 1), cx1 = min(max(x0 + 1, 0), W_ - 1);
        const int i00 = cy0 * W_ + cx0, i01 = cy0 * W_ + cx1;
        const int i10 = cy1 * W_ + cx0, i11 = cy1 * W_ + cx1;
        const float m00 = (vy0 & vx0) ? (1.f - ly) * (1.f - lx) * mk : 0.f;
        const float m01 = (vy0 & vx1) ? (1.f - ly) * lx         * mk : 0.f;
        const float m10 = (vy1 & vx0) ? ly * (1.f - lx)         * mk : 0.f;
        const float m11 = (vy1 & vx1) ? ly * lx                 * mk : 0.f;
        for (int ci = cig * 32; ci < cig * 32 + 32; ++ci) {
            const float* xc = xb + ci * HW;
            float v = m00 * reluf(xc[i00]) + m01 * reluf(xc[i01])
                    + m10 * reluf(xc[i10]) + m11 * reluf(xc[i11]);
            col[pix * LDS_STRIDE + ci * 9 + k] = v;
        }
    }
    __syncthreads();

    // ---- WMMA GEMM: wave owns co rows [16*wave, 16*wave+16), all 64 pixels ----
    const int wave = t >> 5, lane = t & 31;
    const int Mb = wave * 16;
    const int lm = lane & 15;
    const int lk = (lane >> 4) * 2;
    const float* arow  = w_dcn + (long)(Mb + lm) * CK + lk;   // [co][ci*9+k]
    const float* bbase = col + lm * LDS_STRIDE + lk;

    v8f acc[4] = {{}, {}, {}, {}};
    v2f a, bf[4];
    a.x = arow[0]; a.y = arow[1];
    #pragma unroll
    for (int j = 0; j < 4; ++j) {
        const float* p = bbase + j * 16 * LDS_STRIDE;
        bf[j].x = p[0]; bf[j].y = p[1];
    }
    for (int kk = 0; kk < CK - 4; kk += 4) {
        v2f an, bn[4];                            // prefetch next K-step
        an.x = arow[kk + 4]; an.y = arow[kk + 5];
        #pragma unroll
        for (int j = 0; j < 4; ++j) {
            const float* p = bbase + j * 16 * LDS_STRIDE + kk + 4;
            bn[j].x = p[0]; bn[j].y = p[1];
        }
        #pragma unroll
        for (int j = 0; j < 4; ++j)
            acc[j] = __builtin_amdgcn_wmma_f32_16x16x4_f32(
                         false, a, false, bf[j], (short)0, acc[j], false, false);
        a = an;
        #pragma unroll
        for (int j = 0; j < 4; ++j) bf[j] = bn[j];
    }
    #pragma unroll
    for (int j = 0; j < 4; ++j)
        acc[j] = __builtin_amdgcn_wmma_f32_16x16x4_f32(
                     false, a, false, bf[j], (short)0, acc[j], false, false);

    const int hi = lane >> 4;
    #pragma unroll
    for (int j = 0; j < 4; ++j) {
        const int nn = nb + j * 16 + lm;
        #pragma unroll
        for (int r = 0; r < 8; ++r) {
            const int co = Mb + r + 8 * hi;
            out_pre[(long)co * N_TOT + nn] = acc[j][r] + b_dcn[co];
        }
    }
}

// ---------------------------------------------------------------------------
// Kernel 3: per-channel mean/var over 51200 elems -> scale/shift for BN.
// ---------------------------------------------------------------------------
extern "C" __global__ void __launch_bounds__(256)
stats_kernel(const float* __restrict__ out_pre, const float* __restrict__ gamma,
             const float* __restrict__ beta, float* __restrict__ ss)
{
    __shared__ float s0[256];
    __shared__ float s1[256];
    const int ch = blockIdx.x, t = threadIdx.x;
    const float* p = out_pre + (long)ch * N_TOT;
    float sum = 0.f, sq = 0.f;
    for (int i = t; i < N_TOT; i += 256) {
        float v = p[i];
        sum += v;
        sq  += v * v;
    }
    s0[t] = sum; s1[t] = sq;
    __syncthreads();
    for (int s = 128; s > 0; s >>= 1) {
        if (t < s) { s0[t] += s0[t + s]; s1[t] += s1[t + s]; }
        __syncthreads();
    }
    if (t == 0) {
        const float mu  = s0[0] / (float)N_TOT;
        const float var = s1[0] / (float)N_TOT - mu * mu;
        const float sc  = gamma[ch] * rsqrtf(var + EPS_);
        ss[ch]        = sc;
        ss[128 + ch]  = beta[ch] - mu * sc;
    }
}

// ---------------------------------------------------------------------------
// Kernel 4: apply BN, write NCHW output.
// ---------------------------------------------------------------------------
extern "C" __global__ void __launch_bounds__(256)
bn_apply_kernel(const float* __restrict__ out_pre, const float* __restrict__ ss,
                float* __restrict__ out)
{
    const int idx = blockIdx.x * 256 + threadIdx.x;   // < B*C*HW exactly
    const int b   = idx / (C_ * HW);
    const int rem = idx % (C_ * HW);
    const int co  = rem / HW;
    const int hw  = rem % HW;
    const float v = out_pre[(long)co * N_TOT + b * HW + hw];
    out[idx] = v * ss[co] + ss[128 + co];
}

// ---------------------------------------------------------------------------
// Host launch
// ---------------------------------------------------------------------------
extern "C" void kernel_launch(void* const* d_in, const int* in_sizes, int n_in,
                              void* d_out, int out_size, void* d_ws, size_t ws_size,
                              hipStream_t stream)
{
    const float* x     = (const float*)d_in[0];
    const float* w_off = (const float*)d_in[1];
    const float* b_off = (const float*)d_in[2];
    const float* w_dcn = (const float*)d_in[3];
    const float* b_dcn = (const float*)d_in[4];
    const float* gamma = (const float*)d_in[5];
    const float* beta  = (const float*)d_in[6];

    float* om      = (float*)d_ws;                       // 27  * 51200 f32
    float* out_pre = om + (size_t)27 * N_TOT;            // 128 * 51200 f32
    float* ss      = out_pre + (size_t)C_ * N_TOT;       // 256 f32
    float* wpad    = ss + 256;                           // 32 * 1152 f32

    (void)hipFuncSetAttribute((const void*)offset_conv_kernel,
                              hipFuncAttributeMaxDynamicSharedMemorySize, LDS_BYTES);
    (void)hipFuncSetAttribute((const void*)dcn_conv_kernel,
                              hipFuncAttributeMaxDynamicSharedMemorySize, LDS_BYTES);

    prep_weights_kernel<<<(32 * CK) / 256, 256, 0, stream>>>(w_off, wpad);
    offset_conv_kernel<<<N_TOT / TILE_N, 256, LDS_BYTES, stream>>>(x, wpad, b_off, om);
    dcn_conv_kernel  <<<N_TOT / TILE_N, 256, LDS_BYTES, stream>>>(x, w_dcn, b_dcn, om, out_pre);
    stats_kernel     <<<C_, 256, 0, stream>>>(out_pre, gamma, beta, ss);
    bn_apply_kernel  <<<(B_ * C_ * HW) / 256, 256, 0, stream>>>(out_pre, ss, (float*)d_out);
}